// RSCUCalculator_19533511262776
// MI455X (gfx1250) — compile-verified
//
#include <hip/hip_runtime.h>
#include <hip/hip_bf16.h>
#include <math.h>

// ---------------------------------------------------------------------------
// RSCU + KL divergence for MI455X (gfx1250, wave32).
//  Kernel 1: per-row masked codon histograms (pred+target) via LDS-privatized
//            per-wave histograms (ds_add atomics), int4-vectorized loads.
//            ~36MB read -> pure HBM-bound (~1.6us at 23.3 TB/s).
//  Kernel 2: per 16 rows, grouped codon->AA sums and AA->codon broadcast are
//            expressed as indicator-matrix matmuls on V_WMMA_F32_16X16X4_F32
//            (exact: all operands are small integers in f32), then the tiny
//            eps/normalize/log/KL epilogue with wave32 shuffle reductions.
// ---------------------------------------------------------------------------

typedef __attribute__((ext_vector_type(2))) float v2f;
typedef __attribute__((ext_vector_type(8))) float v8f;

#define NC   65   // real codon count
#define NCP  80   // padded codon dim (multiple of 16)
#define NAAP 32   // padded AA dim (multiple of 16; real = 24)

// codon -> amino acid (contiguous runs per the reference's _SIZES), padded with -1
__constant__ int C2AA[NCP] = {
  0,
  3,3,3,3,
  4,4,4,4,4,4,
  5,5, 6,6, 7,7, 8,8, 9,9,
  10,10,10,
  11,11,
  12,12,12,12,12,12,
  13,
  14,14,
  15,15,15,15,
  16,16,
  17,17,17,17,17,17,
  18,18,18,18,
  19,19,19,19,
  20,
  21,21,
  22,22,22,22,
  23,23,23,
  -1,-1,-1,-1,-1,-1,-1,-1,-1,-1,-1,-1,-1,-1,-1
};

// NSYN_CODON: _NSYN[3:23] = sizes[:20]  => AA 23 (last 3 codons) gets 0.
__constant__ float NSYN[NCP] = {
  0.f,
  4.f,4.f,4.f,4.f,
  6.f,6.f,6.f,6.f,6.f,6.f,
  2.f,2.f, 2.f,2.f, 2.f,2.f, 2.f,2.f, 2.f,2.f,
  3.f,3.f,3.f,
  2.f,2.f,
  6.f,6.f,6.f,6.f,6.f,6.f,
  1.f,
  2.f,2.f,
  4.f,4.f,4.f,4.f,
  2.f,2.f,
  6.f,6.f,6.f,6.f,6.f,6.f,
  4.f,4.f,4.f,4.f,
  4.f,4.f,4.f,4.f,
  1.f,
  2.f,2.f,
  4.f,4.f,4.f,4.f,
  0.f,0.f,0.f,
  0.f,0.f,0.f,0.f,0.f,0.f,0.f,0.f,0.f,0.f,0.f,0.f,0.f,0.f,0.f
};

// ---------------------------------------------------------------------------
// Kernel 1: histograms. One block (256 threads = 8 waves) per batch row.
// ---------------------------------------------------------------------------
__global__ void rscu_hist_kernel(const int* __restrict__ pred,
                                 const int* __restrict__ targ,
                                 const unsigned char* __restrict__ mask, // JAX bool = 1 byte
                                 float* __restrict__ wsP,
                                 float* __restrict__ wsT,
                                 int L)
{
  constexpr int NW = 8;   // waves per block (256 threads, wave32)
  constexpr int HB = 72;  // per-wave bins, padded to dodge some bank aliasing
  __shared__ unsigned int h[NW * 2 * HB];

  const int row  = blockIdx.x;
  const int tid  = threadIdx.x;
  const int wave = tid >> 5;
  unsigned int* hp = &h[(wave * 2 + 0) * HB];
  unsigned int* ht = &h[(wave * 2 + 1) * HB];

  for (int i = tid; i < NW * 2 * HB; i += blockDim.x) h[i] = 0u;
  __syncthreads();

  const int4*        p4 = (const int4*)(pred + (size_t)row * L);
  const int4*        t4 = (const int4*)(targ + (size_t)row * L);
  const unsigned int* m4 = (const unsigned int*)(mask + (size_t)row * L);
  const int n4 = L >> 2;

  for (int j = tid; j < n4; j += blockDim.x) {
    int4 pv = p4[j];
    int4 tv = t4[j];
    unsigned int mv = m4[j];
    if (mv & 0x000000FFu) {
      if ((unsigned)(pv.x - 1) < 64u) atomicAdd(&hp[pv.x], 1u);
      if ((unsigned)(tv.x - 1) < 64u) atomicAdd(&ht[tv.x], 1u);
    }
    if (mv & 0x0000FF00u) {
      if ((unsigned)(pv.y - 1) < 64u) atomicAdd(&hp[pv.y], 1u);
      if ((unsigned)(tv.y - 1) < 64u) atomicAdd(&ht[tv.y], 1u);
    }
    if (mv & 0x00FF0000u) {
      if ((unsigned)(pv.z - 1) < 64u) atomicAdd(&hp[pv.z], 1u);
      if ((unsigned)(tv.z - 1) < 64u) atomicAdd(&ht[tv.z], 1u);
    }
    if (mv & 0xFF000000u) {
      if ((unsigned)(pv.w - 1) < 64u) atomicAdd(&hp[pv.w], 1u);
      if ((unsigned)(tv.w - 1) < 64u) atomicAdd(&ht[tv.w], 1u);
    }
  }
  // scalar tail (empty when L % 4 == 0, as in the reference: L = 8192)
  const int* ps = pred + (size_t)row * L;
  const int* ts = targ + (size_t)row * L;
  const unsigned char* ms = mask + (size_t)row * L;
  for (int j = (n4 << 2) + tid; j < L; j += blockDim.x) {
    if (ms[j]) {
      int pv = ps[j], tv = ts[j];
      if ((unsigned)(pv - 1) < 64u) atomicAdd(&hp[pv], 1u);
      if ((unsigned)(tv - 1) < 64u) atomicAdd(&ht[tv], 1u);
    }
  }
  __syncthreads();

  // reduce wave-private histograms, write padded float rows to workspace
  for (int c = tid; c < NCP; c += blockDim.x) {
    unsigned int sp = 0u, st = 0u;
    if (c < NC) {
      #pragma unroll
      for (int w = 0; w < NW; ++w) {
        sp += h[(w * 2 + 0) * HB + c];
        st += h[(w * 2 + 1) * HB + c];
      }
    }
    wsP[(size_t)row * NCP + c] = (float)sp;
    wsT[(size_t)row * NCP + c] = (float)st;
  }
}

// ---------------------------------------------------------------------------
// Kernel 2: WMMA grouped sums + KL epilogue. One wave32 per 16 batch rows.
// ---------------------------------------------------------------------------
__device__ __forceinline__ float wave_sum32(float v) {
  #pragma unroll
  for (int off = 16; off > 0; off >>= 1) v += __shfl_xor(v, off, 32);
  return v;
}

__global__ void rscu_finalize_kernel(const float* __restrict__ wsP,
                                     const float* __restrict__ wsT,
                                     const int* __restrict__ species,
                                     const float* __restrict__ refd,
                                     float* __restrict__ out,
                                     int S, int B)
{
  __shared__ float aaP[16 * NAAP];
  __shared__ float aaT[16 * NAAP];
  __shared__ float totP[16 * NCP];
  __shared__ float totT[16 * NCP];

  const int lane = threadIdx.x;          // 32 threads, EXEC all ones (WMMA req.)
  const int row0 = blockIdx.x * 16;
  const int m    = lane & 15;            // A-matrix row owned by this lane
  const int koff = (lane >> 4) << 1;     // lanes 16-31 hold K+2/K+3 in VGPR0/1
  const int ncol = lane & 15;            // B/D column owned by this lane
  const int mrow = (lane >> 4) << 3;     // D layout: lanes 16-31 hold M=i+8
  const int rowm = min(row0 + m, B - 1);

  // ---- Part 1: aa_tot(16x32) = counts(16x80) @ G(80x32),  G[c,a]=[C2AA[c]==a]
  v8f accP0 = {}; v8f accP1 = {}; v8f accT0 = {}; v8f accT1 = {};
  #pragma unroll
  for (int kk = 0; kk < NCP / 4; ++kk) {
    const int k0 = kk * 4 + koff;
    v2f ap, at, b0, b1;
    ap.x = wsP[(size_t)rowm * NCP + k0];
    ap.y = wsP[(size_t)rowm * NCP + k0 + 1];
    at.x = wsT[(size_t)rowm * NCP + k0];
    at.y = wsT[(size_t)rowm * NCP + k0 + 1];
    const int a0 = C2AA[k0], a1 = C2AA[k0 + 1];   // -1 on padded codons
    b0.x = (a0 == ncol)      ? 1.f : 0.f;
    b0.y = (a1 == ncol)      ? 1.f : 0.f;
    b1.x = (a0 == ncol + 16) ? 1.f : 0.f;
    b1.y = (a1 == ncol + 16) ? 1.f : 0.f;
    accP0 = __builtin_amdgcn_wmma_f32_16x16x4_f32(false, ap, false, b0, (short)0, accP0, false, false);
    accP1 = __builtin_amdgcn_wmma_f32_16x16x4_f32(false, ap, false, b1, (short)0, accP1, false, false);
    accT0 = __builtin_amdgcn_wmma_f32_16x16x4_f32(false, at, false, b0, (short)0, accT0, false, false);
    accT1 = __builtin_amdgcn_wmma_f32_16x16x4_f32(false, at, false, b1, (short)0, accT1, false, false);
  }
  // spill D tiles (C/D layout) to LDS as aa[m][a]
  #pragma unroll
  for (int i = 0; i < 8; ++i) {
    aaP[(mrow + i) * NAAP + ncol]      = accP0[i];
    aaP[(mrow + i) * NAAP + 16 + ncol] = accP1[i];
    aaT[(mrow + i) * NAAP + ncol]      = accT0[i];
    aaT[(mrow + i) * NAAP + 16 + ncol] = accT1[i];
  }
  __syncthreads();

  // ---- Part 2: tot_c(16x80) = aa_tot(16x32) @ G^T(32x80)
  v8f zacc = {};
  v8f tp[5], tt[5];
  #pragma unroll
  for (int nt = 0; nt < 5; ++nt) { tp[nt] = zacc; tt[nt] = zacc; }
  #pragma unroll
  for (int kk = 0; kk < NAAP / 4; ++kk) {
    const int k0 = kk * 4 + koff;
    v2f ap, at;
    ap.x = aaP[m * NAAP + k0];  ap.y = aaP[m * NAAP + k0 + 1];
    at.x = aaT[m * NAAP + k0];  at.y = aaT[m * NAAP + k0 + 1];
    #pragma unroll
    for (int nt = 0; nt < 5; ++nt) {
      const int n   = nt * 16 + ncol;   // codon column
      const int aan = C2AA[n];          // -1 on padded codons
      v2f b;
      b.x = (aan == k0)     ? 1.f : 0.f;
      b.y = (aan == k0 + 1) ? 1.f : 0.f;
      tp[nt] = __builtin_amdgcn_wmma_f32_16x16x4_f32(false, ap, false, b, (short)0, tp[nt], false, false);
      tt[nt] = __builtin_amdgcn_wmma_f32_16x16x4_f32(false, at, false, b, (short)0, tt[nt], false, false);
    }
  }
  #pragma unroll
  for (int nt = 0; nt < 5; ++nt) {
    #pragma unroll
    for (int i = 0; i < 8; ++i) {
      totP[(mrow + i) * NCP + nt * 16 + ncol] = tp[nt][i];
      totT[(mrow + i) * NCP + nt * 16 + ncol] = tt[nt][i];
    }
  }
  __syncthreads();

  // ---- Epilogue: RSCU -> blend with species reference -> normalize -> KL
  for (int mm = 0; mm < 16; ++mm) {
    const int row = row0 + mm;
    if (row >= B) break;
    const int  sid = species[row];
    const bool inr = (sid >= 0) && (sid < S);
    const float* rowP = wsP + (size_t)row * NCP;
    const float* rowT = wsT + (size_t)row * NCP;

    float pr[3], tr[3];
    float sp = 0.f, st = 0.f;
    #pragma unroll
    for (int it = 0; it < 3; ++it) {
      const int c = lane + it * 32;
      float p_raw = 0.f, t_raw = 0.f;
      if (c < NC) {
        const float pc = rowP[c];
        const float tc = rowT[c];
        const float tpv = totP[mm * NCP + c];
        const float ttv = totT[mm * NCP + c];
        const float ns  = NSYN[c];
        const float prscu = (tpv > 0.f) ? pc * ns / fmaxf(tpv, 1.f) : 0.f;
        const float trscu = (ttv > 0.f) ? tc * ns / fmaxf(ttv, 1.f) : 0.f;
        const float rf = inr ? refd[(size_t)sid * NC + c] : 0.f;
        p_raw = prscu + 1e-8f;
        t_raw = 0.7f * trscu + 0.3f * rf + 1e-8f;
      }
      pr[it] = p_raw; tr[it] = t_raw;
      sp += p_raw; st += t_raw;
    }
    sp = wave_sum32(sp);
    st = wave_sum32(st);

    float kl = 0.f;
    #pragma unroll
    for (int it = 0; it < 3; ++it) {
      if (pr[it] > 0.f) {               // only real codons had eps added
        const float pn = pr[it] / sp;
        const float tn = tr[it] / st;
        kl += tn * __logf(tn / pn);
      }
    }
    kl = wave_sum32(kl);
    if (lane == 0) out[row] = kl;
  }
}

// ---------------------------------------------------------------------------
extern "C" void kernel_launch(void* const* d_in, const int* in_sizes, int n_in,
                              void* d_out, int out_size, void* d_ws, size_t ws_size,
                              hipStream_t stream) {
  (void)n_in; (void)out_size; (void)ws_size;
  const int*           pred    = (const int*)d_in[0];            // (B,L) int32
  const int*           targ    = (const int*)d_in[1];            // (B,L) int32
  /* d_in[2] = aa_ids: unused (derivable from target ids) */
  const int*           species = (const int*)d_in[3];            // (B,) int32
  const unsigned char* mask    = (const unsigned char*)d_in[4];  // (B,L) bool (1B)
  const float*         refd    = (const float*)d_in[5];          // (S,65) f32

  const int B = in_sizes[3];
  const int L = in_sizes[0] / B;
  const int S = in_sizes[5] / NC;

  float* wsP = (float*)d_ws;                 // (B,80) padded pred counts
  float* wsT = wsP + (size_t)B * NCP;        // (B,80) padded target counts
  float* out = (float*)d_out;                // (B,) kl

  rscu_hist_kernel<<<B, 256, 0, stream>>>(pred, targ, mask, wsP, wsT, L);

  const int nBlocks = (B + 15) / 16;
  rscu_finalize_kernel<<<nBlocks, 32, 0, stream>>>(wsP, wsT, species, refd, out, S, B);
}